// Model_57595511439941
// MI455X (gfx1250) — compile-verified
//
#include <hip/hip_runtime.h>
#include <math.h>

// ---------------- problem constants ----------------
#define KLEN  1023
#define KPAD  1024      // K padded with a zero row for branchless WMMA loops
#define HN    16
#define EN    256
#define TN    8
#define BN    8
#define LLEN  8184
#define DFF   512
#define LOUT  7162      // VALID conv output length
#define CHUNK 896       // LOUT padded to 7168 = 8 chunks of 896 (multiple of 16)
#define NROWS 128       // B*H
#define NF    128       // number of recon filters
#define EPSF  1.1920929e-07f

// ---------------- workspace layout (float offsets) ----------------
#define Z_OFF      0                        // 128*1023 floats
#define CODES_OFF  (NROWS*KLEN)             // 128 ints (punned)
#define ACC_OFF    131072
#define COUNTS_OFF (ACC_OFF)                // 256
#define VQ_OFF     (ACC_OFF+256)            // 1
#define RECON_OFF  (ACC_OFF+257)            // 1
#define PSUM_OFF   (ACC_OFF+258)            // 8*256
#define FEAT_OFF   (ACC_OFF+258+2048)       // 8*256
#define CMAX_OFF   (ACC_OFF+258+4096)       // 8*8*256
#define ACC_ZEROS  (258+4096)               // 4354 zeroed floats
#define CMAX_N     (BN*TN*EN)               // 16384 -inf floats
#define EMBT_OFF   151936                   // k-major embedding: 1024*256
#define WQT_OFF    (EMBT_OFF + KPAD*EN)     // k-major quantized filters: 1024*128
// total ws: WQT_OFF + KPAD*NF = 545,152 floats ~= 2.2 MB

typedef __attribute__((ext_vector_type(2))) float v2f;
typedef __attribute__((ext_vector_type(8))) float v8f;

__device__ inline void atomicMaxFloat(float* addr, float val) {
  // sign-split trick: int max for >=0, uint min for <0 ; init must be -inf
  if (val >= 0.0f) atomicMax((int*)addr, __float_as_int(val));
  else             atomicMin((unsigned int*)addr, __float_as_uint(val));
}

// ---------------- init: zero accumulators, -inf chunk maxes ----------------
__global__ void initK(float* ws) {
  int t = blockIdx.x * blockDim.x + threadIdx.x;
  if (t < ACC_ZEROS)                ws[ACC_OFF + t] = 0.0f;
  else if (t < ACC_ZEROS + CMAX_N)  ws[CMAX_OFF + (t - ACC_ZEROS)] = -__builtin_inff();
}

// ---------------- k-major transpose of embedding, zero pad row k=1023 ------
__global__ void transEmbK(const float* __restrict__ emb, float* __restrict__ ws) {
  int t = blockIdx.x * 256 + threadIdx.x;    // 1024*256 threads
  int k = t >> 8, e = t & 255;
  ws[EMBT_OFF + t] = (k < KLEN) ? emb[e * KLEN + k] : 0.0f;
}

// ---------------- k-major gather of quantized filters (after vqK) ----------
__global__ void transWqK(const float* __restrict__ emb, float* __restrict__ ws) {
  int t = blockIdx.x * 256 + threadIdx.x;    // 1024*128 threads
  int k = t >> 7, f = t & 127;
  int code = ((const int*)ws)[CODES_OFF + f];
  ws[WQT_OFF + t] = (k < KLEN) ? emb[code * KLEN + k] : 0.0f;
}

// ---------------- frames + Hann window + circular autocov (ifftshifted) ----
__global__ void autocovK(const float* __restrict__ X, float* __restrict__ ws) {
  __shared__ float f[KLEN];
  __shared__ float red[256];
  int row = blockIdx.x;            // b*H + h
  int b = row / HN, h = row % HN;
  int tid = threadIdx.x;
  int pos = (int)(6137.0f * (float)h / 15.0f);   // linspace(0,end,16) trunc
  const float* xb = X + b * LLEN + pos;
  for (int n = tid; n < KLEN; n += 256) f[n] = xb[n];
  __syncthreads();
  float m = 0.0f;
  for (int n = tid; n < KLEN; n += 256) m = fmaxf(m, fabsf(f[n]));
  red[tid] = m; __syncthreads();
  for (int s = 128; s > 0; s >>= 1) { if (tid < s) red[tid] = fmaxf(red[tid], red[tid + s]); __syncthreads(); }
  float fmax = red[0]; if (fmax == 0.0f) fmax = EPSF;
  float inv = 1.0f / fmax;
  __syncthreads();
  const float w0 = 6.283185307179586f / (float)KLEN;
  for (int n = tid; n < KLEN; n += 256) {
    float win = 0.5f * (1.0f - __cosf(w0 * (float)n));
    f[n] = f[n] * win * inv;                      // elementwise, in place
  }
  __syncthreads();
  float* Zrow = ws + Z_OFF + row * KLEN;
  for (int k = tid; k < KLEN; k += 256) {
    float r = 0.0f;
    for (int n = 0; n < KLEN - k; ++n) r = fmaf(f[n], f[n + k], r);
    for (int n = KLEN - k; n < KLEN; ++n) r = fmaf(f[n], f[n + k - KLEN], r);
    Zrow[(k + 512) % KLEN] = r;                   // ifftshift (n odd: roll +512)
  }
}

// ---------------- VQ: argmin over 256 codes + histogram + vq sq-err --------
__global__ void vqK(const float* __restrict__ emb, float* __restrict__ ws) {
  __shared__ float z[KLEN];
  __shared__ float sc[256];
  __shared__ int   si[256];
  __shared__ int   codeSh;
  int row = blockIdx.x, tid = threadIdx.x;
  const float* Zrow = ws + Z_OFF + row * KLEN;
  for (int k = tid; k < KLEN; k += 256) z[k] = Zrow[k];
  __syncthreads();
  const float* e = emb + tid * KLEN;
  float ee = 0.0f, dot = 0.0f;
  for (int k = 0; k < KLEN; ++k) { float ev = e[k]; ee = fmaf(ev, ev, ee); dot = fmaf(ev, z[k], dot); }
  sc[tid] = ee - 2.0f * dot; si[tid] = tid; __syncthreads();
  for (int s = 128; s > 0; s >>= 1) {
    if (tid < s) {
      float a = sc[tid], bb = sc[tid + s];
      if (bb < a || (bb == a && si[tid + s] < si[tid])) { sc[tid] = bb; si[tid] = si[tid + s]; }
    }
    __syncthreads();
  }
  if (tid == 0) {
    int c = si[0];
    ((int*)ws)[CODES_OFF + row] = c;
    atomicAdd(ws + COUNTS_OFF + c, 1.0f);
    codeSh = c;
  }
  __syncthreads();
  const float* eq = emb + codeSh * KLEN;
  float s2 = 0.0f;
  for (int k = tid; k < KLEN; k += 256) { float d = eq[k] - z[k]; s2 = fmaf(d, d, s2); }
  sc[tid] = s2; __syncthreads();
  for (int s = 128; s > 0; s >>= 1) { if (tid < s) sc[tid] += sc[tid + s]; __syncthreads(); }
  if (tid == 0) atomicAdd(ws + VQ_OFF, sc[0]);
}

// ---------------- recon conv (SAME, 128 quantized filters) via WMMA f32 ----
// out[(i,l), f] = sum_k X[i, l+k-511] * WqT[k][f]; loss vs X accumulated.
// Branchless: WqT is k-major, zero-padded to 1024; B loads lane-consecutive.
__global__ void reconK(const float* __restrict__ X, float* __restrict__ ws) {
  __shared__ float xs[1040];
  __shared__ float red[256];
  int l0 = blockIdx.x * 16;
  int i  = blockIdx.y;
  int tid = threadIdx.x;
  for (int j = tid; j < 1040; j += 256) {
    int gl = l0 - 511 + j;
    xs[j] = (gl >= 0 && gl < LLEN) ? X[i * LLEN + gl] : 0.0f;
  }
  __syncthreads();
  int wid = tid >> 5, lane = tid & 31;
  int n   = lane & 15;          // A-row m and B-col (filter) index share lane&15
  int hi  = lane >> 4;          // lanes 16-31 carry K+2/K+3
  int k0o = hi ? 2 : 0, k1o = hi ? 3 : 1;
  const float* BT = ws + WQT_OFF + wid * 16 + n;   // column f0+n, stride NF
  v8f c0 = {0.f,0.f,0.f,0.f,0.f,0.f,0.f,0.f};
  v8f c1 = {0.f,0.f,0.f,0.f,0.f,0.f,0.f,0.f};
  for (int kk = 0; kk < KPAD / 2; kk += 4) {       // two independent chains
    v2f a0, b0, a1, b1;
    a0.x = xs[n + kk + k0o];
    a0.y = xs[n + kk + k1o];
    b0.x = BT[(kk + k0o) * NF];
    b0.y = BT[(kk + k1o) * NF];
    a1.x = xs[n + 512 + kk + k0o];
    a1.y = xs[n + 512 + kk + k1o];
    b1.x = BT[(512 + kk + k0o) * NF];
    b1.y = BT[(512 + kk + k1o) * NF];
    c0 = __builtin_amdgcn_wmma_f32_16x16x4_f32(false, a0, false, b0, (short)0, c0, false, false);
    c1 = __builtin_amdgcn_wmma_f32_16x16x4_f32(false, a1, false, b1, (short)0, c1, false, false);
  }
  float acc = 0.0f;
  for (int r = 0; r < 8; ++r) {                    // C layout: M = r + hi*8, N = lane&15
    int mloc = r + hi * 8;
    int l = l0 + mloc;
    if (l < LLEN) { float d = (c0[r] + c1[r]) - xs[mloc + 511]; acc = fmaf(d, d, acc); }
  }
  red[tid] = acc; __syncthreads();
  for (int s = 128; s > 0; s >>= 1) { if (tid < s) red[tid] += red[tid + s]; __syncthreads(); }
  if (tid == 0) atomicAdd(ws + RECON_OFF, red[0]);
}

// ---------------- projection conv (VALID, all 256 codes) via WMMA f32 ------
// proj[(i,l), e] = sum_k X[i, l+k] * embT[k][e]; per-(i,e) sum + per-chunk max.
__global__ void projK(const float* __restrict__ X, float* __restrict__ ws) {
  __shared__ float xs[1040];
  int l0 = blockIdx.x * 16;
  int i  = blockIdx.y;
  int tid = threadIdx.x;
  for (int j = tid; j < 1040; j += 256) {
    int gl = l0 + j;
    xs[j] = (gl < LLEN) ? X[i * LLEN + gl] : 0.0f;
  }
  __syncthreads();
  int wid = tid >> 5, lane = tid & 31;
  int n = lane & 15, hi = lane >> 4;
  int k0o = hi ? 2 : 0, k1o = hi ? 3 : 1;
  int t = l0 / CHUNK;                              // 896 % 16 == 0: tile in one chunk
  for (int pass = 0; pass < 2; ++pass) {
    int e0 = wid * 16 + pass * 128;
    const float* BT = ws + EMBT_OFF + e0 + n;      // column e0+n, stride EN
    v8f c0 = {0.f,0.f,0.f,0.f,0.f,0.f,0.f,0.f};
    v8f c1 = {0.f,0.f,0.f,0.f,0.f,0.f,0.f,0.f};
    for (int kk = 0; kk < KPAD / 2; kk += 4) {
      v2f a0, b0, a1, b1;
      a0.x = xs[n + kk + k0o];
      a0.y = xs[n + kk + k1o];
      b0.x = BT[(kk + k0o) * EN];
      b0.y = BT[(kk + k1o) * EN];
      a1.x = xs[n + 512 + kk + k0o];
      a1.y = xs[n + 512 + kk + k1o];
      b1.x = BT[(512 + kk + k0o) * EN];
      b1.y = BT[(512 + kk + k1o) * EN];
      c0 = __builtin_amdgcn_wmma_f32_16x16x4_f32(false, a0, false, b0, (short)0, c0, false, false);
      c1 = __builtin_amdgcn_wmma_f32_16x16x4_f32(false, a1, false, b1, (short)0, c1, false, false);
    }
    float sum = 0.0f, mx = -__builtin_inff(); int cnt = 0;
    for (int r = 0; r < 8; ++r) {
      int l = l0 + r + hi * 8;
      float v = c0[r] + c1[r];
      if (l < LOUT) { sum += v; mx = fmaxf(mx, v); cnt++; }
    }
    if (cnt) {
      atomicAdd(ws + PSUM_OFF + i * EN + e0 + n, sum);
      atomicMaxFloat(ws + CMAX_OFF + (i * TN + t) * EN + e0 + n, mx);
    }
  }
}

// ---------------- pooled x -> LayerNorm -> FFN -> feat (mean over T) -------
__global__ void mlpK(const float* __restrict__ w1, const float* __restrict__ b1,
                     const float* __restrict__ w2, const float* __restrict__ b2,
                     const float* __restrict__ g,  const float* __restrict__ beta,
                     float* __restrict__ ws) {
  __shared__ float xn[EN];
  __shared__ float hidv[DFF];
  __shared__ float red[256];
  int i = blockIdx.x, t = blockIdx.y, tid = threadIdx.x;
  float v = ws[CMAX_OFF + (i * TN + t) * EN + tid];
  if (t == TN - 1) {                                // mean-padded tail positions
    float mean = ws[PSUM_OFF + i * EN + tid] / (float)LOUT;
    v = fmaxf(v, mean);
  }
  red[tid] = v; __syncthreads();
  for (int s = 128; s > 0; s >>= 1) { if (tid < s) red[tid] += red[tid + s]; __syncthreads(); }
  float mu = red[0] / (float)EN; __syncthreads();
  float d = v - mu;
  red[tid] = d * d; __syncthreads();
  for (int s = 128; s > 0; s >>= 1) { if (tid < s) red[tid] += red[tid + s]; __syncthreads(); }
  float var = red[0] / (float)EN;
  xn[tid] = d * rsqrtf(var + 1e-5f) * g[tid] + beta[tid];
  __syncthreads();
  for (int j = tid; j < DFF; j += 256) {
    float hsum = b1[j];
    for (int e2 = 0; e2 < EN; ++e2) hsum = fmaf(xn[e2], w1[e2 * DFF + j], hsum);
    hidv[j] = fmaxf(hsum, 0.0f);
  }
  __syncthreads();
  float o = b2[tid];
  for (int j = 0; j < DFF; ++j) o = fmaf(hidv[j], w2[j * EN + tid], o);
  atomicAdd(ws + FEAT_OFF + i * EN + tid, o * (1.0f / (float)TN));
}

// ---------------- classifier + scalar losses ----------------
__global__ void finalK(const float* __restrict__ cw, const float* __restrict__ cb,
                       const float* __restrict__ ws, float* __restrict__ out) {
  int tid = threadIdx.x;
  if (tid < 80) {
    int i = tid / 10, c = tid % 10;
    float s = cb[c];
    const float* f = ws + FEAT_OFF + i * EN;
    for (int e = 0; e < EN; ++e) s = fmaf(f[e], cw[e * 10 + c], s);
    out[tid] = s;
  } else if (tid == 80) {
    out[80] = 1.25f * ws[VQ_OFF] / (float)(NROWS * KLEN);   // q_loss + 0.25*e_loss
  } else if (tid == 81) {
    out[81] = ws[RECON_OFF] / (float)(BN * HN * LLEN);
  } else if (tid == 82) {
    float ent = 0.0f;
    for (int e = 0; e < EN; ++e) {
      float avg = ws[COUNTS_OFF + e] / (float)NROWS;
      ent -= avg * __logf(avg + EPSF);
    }
    out[82] = __expf(ent);
  }
}

extern "C" void kernel_launch(void* const* d_in, const int* in_sizes, int n_in,
                              void* d_out, int out_size, void* d_ws, size_t ws_size,
                              hipStream_t stream) {
  const float* X   = (const float*)d_in[0];
  const float* emb = (const float*)d_in[1];
  const float* w1  = (const float*)d_in[2];
  const float* b1  = (const float*)d_in[3];
  const float* w2  = (const float*)d_in[4];
  const float* b2  = (const float*)d_in[5];
  const float* g   = (const float*)d_in[6];
  const float* be  = (const float*)d_in[7];
  const float* cw  = (const float*)d_in[8];
  const float* cb  = (const float*)d_in[9];
  float* ws  = (float*)d_ws;
  float* out = (float*)d_out;

  initK    <<<(ACC_ZEROS + CMAX_N + 255) / 256, 256, 0, stream>>>(ws);
  transEmbK<<<(KPAD * EN) / 256, 256, 0, stream>>>(emb, ws);
  autocovK <<<NROWS, 256, 0, stream>>>(X, ws);
  vqK      <<<NROWS, 256, 0, stream>>>(emb, ws);
  transWqK <<<(KPAD * NF) / 256, 256, 0, stream>>>(emb, ws);
  reconK   <<<dim3(512, BN), 256, 0, stream>>>(X, ws);     // 512 l-tiles of 16
  projK    <<<dim3(448, BN), 256, 0, stream>>>(X, ws);     // ceil(7162/16)=448
  mlpK     <<<dim3(BN, TN), 256, 0, stream>>>(w1, b1, w2, b2, g, be, ws);
  finalK   <<<1, 256, 0, stream>>>(cw, cb, ws, out);
}